// DeepPolyReLu_59047210385772
// MI455X (gfx1250) — compile-verified
//
#include <hip/hip_runtime.h>

// ---------------------------------------------------------------------------
// DeepPoly ReLU back-substitution, reduced to two fused GEMVs over W:
//   S1 = W @ c,  S2 = |W| @ r   (c,r = input box center/radius)
//   new_ub = us*(S1+S2+b) + ubias ;  new_lb = ls*(S1-S2+b)
// HBM-bound: one 128 MiB pass of W (L2-resident on graph replays).
// MACs on the matrix pipe (V_WMMA_F32_16X16X4_F32); W tiles streamed into
// wave-private LDS with double-buffered GLOBAL_LOAD_ASYNC_TO_LDS_B128.
// ---------------------------------------------------------------------------

typedef float v2f __attribute__((ext_vector_type(2)));
typedef float v8f __attribute__((ext_vector_type(8)));

#define N_ROWS 8192
#define D_IN   4096
#define KSPLIT 4
#define KSEG   (D_IN / KSPLIT)   // 1024 K per wave
#define KTILE  32                // K staged per LDS tile
#define NTILES (KSEG / KTILE)    // 32
#define TSTRIDE 36               // 32 + 4 pad dwords -> conflict-free A gather
#define WAVES_PER_BLOCK 8
#define BLOCK_THREADS   (WAVES_PER_BLOCK * 32)

#if defined(__has_builtin)
#if __has_builtin(__builtin_amdgcn_wmma_f32_16x16x4_f32)
#define HAVE_WMMA_F32X4 1
#endif
#endif

__device__ __forceinline__ v8f wmma_f32x4(v2f a, v2f b, v8f c) {
#ifdef HAVE_WMMA_F32X4
  return __builtin_amdgcn_wmma_f32_16x16x4_f32(false, a, false, b, (short)0, c,
                                               false, false);
#else
  c[0] += a.x * b.x + a.y * b.y;  // keeps compilation alive; wmma==0 flags it
  return c;
#endif
}

// ---------------------------------------------------------------------------
// Prep: interleave box center/radius pairs so one b128 load fetches both
// B-operand pairs of a WMMA K-step: Z[4p..4p+3] = {c2p, c2p+1, r2p, r2p+1}.
// ---------------------------------------------------------------------------
__global__ __launch_bounds__(256) void deeppoly_prep(
    const float* __restrict__ orig_ub, const float* __restrict__ orig_lb,
    float* __restrict__ Z) {
  const int p = blockIdx.x * blockDim.x + threadIdx.x;
  if (p >= D_IN / 2) return;
  const float u0 = orig_ub[2 * p], u1 = orig_ub[2 * p + 1];
  const float l0 = orig_lb[2 * p], l1 = orig_lb[2 * p + 1];
  float4 z;
  z.x = 0.5f * (u0 + l0);
  z.y = 0.5f * (u1 + l1);
  z.z = 0.5f * (u0 - l0);
  z.w = 0.5f * (u1 - l1);
  reinterpret_cast<float4*>(Z)[p] = z;
}

// ---------------------------------------------------------------------------
// Main: each wave = 16 rows x 1024-wide K segment (2048 waves total).
// Barrier-free; LDS tiles are wave-private and double-buffered via async DMA.
// ---------------------------------------------------------------------------
__global__ __launch_bounds__(BLOCK_THREADS) void deeppoly_gemv_wmma(
    const float* __restrict__ W, const float* __restrict__ Z,
    float* __restrict__ s1_out, float* __restrict__ s2_out) {
  __shared__ __align__(16) float tiles[WAVES_PER_BLOCK][2][16 * TSTRIDE];

  const int tid  = threadIdx.x;
  const int wave = tid >> 5;
  const int lane = tid & 31;
  const int gwave   = blockIdx.x * WAVES_PER_BLOCK + wave;  // 0..2047
  const int rowTile = gwave >> 2;                           // 0..511
  const int kspl    = gwave & (KSPLIT - 1);
  const int rowBase = rowTile * 16;
  const int kBase   = kspl * KSEG;

  // A-operand gather per ISA 32-bit A 16x4 layout:
  //   lanes 0-15 : (M=lane,   K=0,1) ; lanes 16-31: (M=lane-16, K=2,3)
  const int halfSel = lane >> 4;
  const int mrow    = lane & 15;
  const int aBase   = mrow * TSTRIDE + (halfSel ? 2 : 0);

  // Coalesced staging map: chunk = j*32+lane -> row chunk>>3, k (chunk&7)*4.
  const int ldRow = lane >> 3;  // + 4*j
  const int ldCol = (lane & 7) * 4;

  const unsigned ldsBuf0 = (unsigned)(uintptr_t)&tiles[wave][0][0];
  const unsigned ldsBuf1 = (unsigned)(uintptr_t)&tiles[wave][1][0];
  const unsigned ldsLane = (unsigned)((ldRow * TSTRIDE + ldCol) * 4);

  const float4* __restrict__ Zf4 = reinterpret_cast<const float4*>(Z);

  v8f accW = {0.f, 0.f, 0.f, 0.f, 0.f, 0.f, 0.f, 0.f};
  v8f accA = {0.f, 0.f, 0.f, 0.f, 0.f, 0.f, 0.f, 0.f};

  // Issue one 16x32 tile (4 x b128 per lane-row group) into LDS buffer `buf`.
  auto issue_tile = [&](int tileIdx, int buf) {
    const unsigned ldsBase = (buf ? ldsBuf1 : ldsBuf0) + ldsLane;
    const float* g0 = W + (size_t)(rowBase + ldRow) * D_IN +
                      (kBase + tileIdx * KTILE + ldCol);
#pragma unroll
    for (int j = 0; j < 4; ++j) {
      const unsigned lds = ldsBase + (unsigned)(4 * TSTRIDE * 4) * j;
      const unsigned long long ga =
          (unsigned long long)(uintptr_t)(g0 + (size_t)(4 * j) * D_IN);
      asm volatile("global_load_async_to_lds_b128 %0, %1, off"
                   :
                   : "v"(lds), "v"(ga)
                   : "memory");
    }
  };

  // Consume tile: 8 WMMA K-steps per flavor (W and |W|).
  auto consume_tile = [&](int tileIdx, int buf) {
    const float* tile = &tiles[wave][buf][0];
    const int pBase = (kBase + tileIdx * KTILE) / 2 + halfSel;
#pragma unroll
    for (int s = 0; s < KTILE / 4; ++s) {
      const v2f a = *reinterpret_cast<const v2f*>(&tile[aBase + 4 * s]);
      v2f aa;
      aa.x = __builtin_fabsf(a.x);
      aa.y = __builtin_fabsf(a.y);
      const float4 z = Zf4[pBase + 2 * s];  // {c0,c1,r0,r1} for this half
      const v2f bc = {z.x, z.y};
      const v2f br = {z.z, z.w};
      accW = wmma_f32x4(a, bc, accW);   // B cols identical -> D cols identical
      accA = wmma_f32x4(aa, br, accA);
    }
  };

  issue_tile(0, 0);
  for (int i = 0; i < NTILES; ++i) {
    if (i + 1 < NTILES) {
      issue_tile(i + 1, (i + 1) & 1);
      // previous tile's 4 async ops complete; next tile's 4 stay in flight
      asm volatile("s_wait_asynccnt 0x4" ::: "memory");
    } else {
      asm volatile("s_wait_asynccnt 0x0" ::: "memory");
    }
    consume_tile(i, i & 1);
  }

  // D layout: VGPR j = row j (lanes 0-15) / row j+8 (lanes 16-31); all 16
  // columns equal -> one predicated lane per row, constant vector indices.
  float* s1w = s1_out + (size_t)kspl * N_ROWS + rowBase;
  float* s2w = s2_out + (size_t)kspl * N_ROWS + rowBase;
#pragma unroll
  for (int j = 0; j < 8; ++j) {
    if (lane == j) {
      s1w[j] = accW[j];
      s2w[j] = accA[j];
    }
    if (lane == 16 + j) {
      s1w[8 + j] = accW[j];
      s2w[8 + j] = accA[j];
    }
  }
}

// ---------------------------------------------------------------------------
// Reduce K-split partials + ReLU-slope epilogue. O(N), negligible.
// ---------------------------------------------------------------------------
__global__ __launch_bounds__(256) void deeppoly_epilogue(
    const float* __restrict__ s1p, const float* __restrict__ s2p,
    const float* __restrict__ prev_ub, const float* __restrict__ prev_lb,
    const float* __restrict__ alpha, const float* __restrict__ bvec,
    float* __restrict__ out) {
  const int i = blockIdx.x * blockDim.x + threadIdx.x;
  if (i >= N_ROWS) return;
  float s1 = 0.f, s2 = 0.f;
#pragma unroll
  for (int s = 0; s < KSPLIT; ++s) {
    s1 += s1p[s * N_ROWS + i];
    s2 += s2p[s * N_ROWS + i];
  }
  const float pu = prev_ub[i], pl = prev_lb[i], al = alpha[i], b = bvec[i];
  const bool neg   = pu <= 0.f;
  const bool cross = (pu > 0.f) && (pl < 0.f);
  const float us    = cross ? pu / (pu - pl) : (neg ? 0.f : 1.f);
  const float ls    = cross ? al : (neg ? 0.f : 1.f);
  const float ubias = cross ? -us * pl : 0.f;
  out[i]          = us * (s1 + s2 + b) + ubias;  // new_ub
  out[N_ROWS + i] = ls * (s1 - s2 + b);          // new_lb
}

// ---------------------------------------------------------------------------
extern "C" void kernel_launch(void* const* d_in, const int* in_sizes, int n_in,
                              void* d_out, int out_size, void* d_ws,
                              size_t ws_size, hipStream_t stream) {
  const float* orig_ub = (const float*)d_in[0];
  const float* orig_lb = (const float*)d_in[1];
  const float* prev_ub = (const float*)d_in[2];
  const float* prev_lb = (const float*)d_in[3];
  const float* alpha   = (const float*)d_in[4];
  const float* W       = (const float*)d_in[5];
  const float* bvec    = (const float*)d_in[6];
  float* out = (float*)d_out;

  // Workspace: S1/S2 K-split partials (2 x 128 KB) + interleaved Z (32 KB).
  float* s1p = (float*)d_ws;
  float* s2p = s1p + KSPLIT * N_ROWS;
  float* Zp  = s2p + KSPLIT * N_ROWS;

  deeppoly_prep<<<(D_IN / 2 + 255) / 256, 256, 0, stream>>>(orig_ub, orig_lb,
                                                            Zp);
  const int nblocks =
      (N_ROWS / 16) * KSPLIT / WAVES_PER_BLOCK;  // 512*4/8 = 256
  deeppoly_gemv_wmma<<<nblocks, BLOCK_THREADS, 0, stream>>>(W, Zp, s1p, s2p);
  deeppoly_epilogue<<<(N_ROWS + 255) / 256, 256, 0, stream>>>(
      s1p, s2p, prev_ub, prev_lb, alpha, bvec, out);
}